// GraphConvolution_21002390077803
// MI455X (gfx1250) — compile-verified
//
#include <hip/hip_runtime.h>
#include <hip/hip_bf16.h>
#include <stdint.h>

// Problem constants (match reference)
#define N_NODES 50000
#define IN_F    256
#define OUT_F   128
#define TILE_M  16
#define EPW     8   // edges per wave in scatter kernel

typedef float v2f  __attribute__((ext_vector_type(2)));
typedef float v8f  __attribute__((ext_vector_type(8)));
typedef unsigned int u32x4 __attribute__((ext_vector_type(4)));
typedef int   i32x4 __attribute__((ext_vector_type(4)));
typedef int   i32x8 __attribute__((ext_vector_type(8)));

// ---------------------------------------------------------------------------
// Kernel 1: h = x @ W^T + b   via V_WMMA_F32_16X16X4_F32
//   block = 256 threads (8 waves). Each block: 16 node rows (TILE_M).
//   Wave w computes the 16x16 tile covering output cols [16w, 16w+16).
//   x tile (16x256 f32 = 16KB) staged into LDS via TDM tensor_load_to_lds
//   (fallback: cooperative float4 copy).
// ---------------------------------------------------------------------------
__global__ void __launch_bounds__(256)
gcn_gemm_wmma(const float* __restrict__ x, const float* __restrict__ W,
              const float* __restrict__ bias, float* __restrict__ h) {
  __shared__ float xt[TILE_M * IN_F];   // at LDS offset 0 (only shared alloc)
  const int m0   = blockIdx.x * TILE_M;
  const int wave = threadIdx.x >> 5;
  const int lane = threadIdx.x & 31;
  const int n0   = wave << 4;

#if __has_builtin(__builtin_amdgcn_tensor_load_to_lds)
  if (threadIdx.x < 32) {
    // Build Tensor DMA Descriptor (D#) per CDNA5 ISA
    uint64_t ga = (uint64_t)(uintptr_t)(x + (size_t)m0 * IN_F); // tile start
    u32x4 g0;
    g0[0] = 1u;                                   // count=1, no gather/restore
    g0[1] = 0u;                                   // lds_addr = 0 (xt base)
    g0[2] = (unsigned)(ga & 0xFFFFFFFFu);         // global_addr[31:0]
    g0[3] = (unsigned)((ga >> 32) & 0x1FFFFFFu)   // global_addr[56:32]
          | (2u << 30);                           // type = 2 ("image")
    i32x8 g1;
    g1[0] = (int)(2u << 16);                      // data_size = 2 -> 4 bytes
    g1[1] = (int)((unsigned)IN_F << 16);          // tensor_dim0[15:0] = 256
    g1[2] = (int)(((unsigned)N_NODES & 0xFFFFu) << 16); // tensor_dim1 lo16
    g1[3] = (int)((unsigned)IN_F << 16);          // tile_dim0 = 256
    g1[4] = TILE_M;                               // tile_dim1 = 16
    g1[5] = IN_F;                                 // tensor_dim0_stride = 256
    g1[6] = 0;
    g1[7] = 0;
    i32x4 g2 = {0, 0, 0, 0};
    i32x4 g3 = {0, 0, 0, 0};
#if __clang_major__ >= 23
    i32x8 gex = {0, 0, 0, 0, 0, 0, 0, 0};
    __builtin_amdgcn_tensor_load_to_lds(g0, g1, g2, g3, gex, 0);
#else
    __builtin_amdgcn_tensor_load_to_lds(g0, g1, g2, g3, 0);
#endif
#if __has_builtin(__builtin_amdgcn_s_wait_tensorcnt)
    __builtin_amdgcn_s_wait_tensorcnt(0);
#else
    asm volatile("s_wait_tensorcnt 0x0" ::: "memory");
#endif
  }
#else
  {
    const float4* src = (const float4*)(x + (size_t)m0 * IN_F);
    float4*       dst = (float4*)xt;
    for (int i = threadIdx.x; i < TILE_M * IN_F / 4; i += 256) dst[i] = src[i];
  }
#endif

#if defined(__AMDGCN__)
  // The TDM writes LDS behind the compiler's back (the builtin only consumes
  // descriptor values). Escape the LDS pointer through an asm block with a
  // memory clobber so LLVM cannot fold the xt loads to undef / delete the
  // LDS allocation.
  {
    void* esc = (void*)&xt[0];
    asm volatile("" : "+v"(esc) : : "memory");
  }
#endif
  __syncthreads();

  // WMMA f32 16x16x4 fragment addressing (ISA §7.12.2, 32-bit A 16x4):
  //   lanes 0-15:  M=lane,    VGPR0=K0, VGPR1=K1
  //   lanes 16-31: M=lane-16, VGPR0=K2, VGPR1=K3
  const int rsel = lane & 15;
  const int koff = (lane >> 4) << 1;   // 0 or 2

  v8f acc;
  {
    float bv = bias[n0 + rsel];        // C/D: col N = lane&15 for every VGPR
#pragma unroll
    for (int i = 0; i < 8; ++i) acc[i] = bv;
  }

  const float* arow = &xt[rsel * IN_F + koff];               // A from LDS
  const float* brow = W + (size_t)(n0 + rsel) * IN_F + koff; // B = W^T col slice

#pragma unroll 8
  for (int k = 0; k < IN_F; k += 4) {
    v2f af = *(const v2f*)(arow + k);   // ds_load_b64
    v2f bf = *(const v2f*)(brow + k);   // global_load_b64 (W hot in L0/L2)
    acc = __builtin_amdgcn_wmma_f32_16x16x4_f32(
        /*neg_a=*/false, af, /*neg_b=*/false, bf,
        /*c_mod=*/(short)0, acc, /*reuse_a=*/false, /*reuse_b=*/false);
  }

  // D layout: VGPR i -> row m0 + i + (lane>=16 ? 8 : 0), col n0 + (lane&15)
  float* hp = h + (size_t)(m0 + ((lane >> 4) << 3)) * OUT_F + n0 + rsel;
#pragma unroll
  for (int i = 0; i < 8; ++i) hp[(size_t)i * OUT_F] = acc[i];
}

// ---------------------------------------------------------------------------
// Kernel 2: zero-init out (harness poisons d_out with 0xAA)
// ---------------------------------------------------------------------------
__global__ void gcn_zero(float* __restrict__ p, int n) {
  int i = (blockIdx.x * blockDim.x + threadIdx.x) * 4;
  if (i + 3 < n) {
    *(float4*)(p + i) = make_float4(0.f, 0.f, 0.f, 0.f);
  } else {
    for (; i < n; ++i) p[i] = 0.f;
  }
}

// ---------------------------------------------------------------------------
// Kernel 3: COO SpMM scatter: out[row[e]] += val[e] * h[col[e]]
//   8 edges per wave32 (one at a time): lane holds 4 contiguous features (float4 = 512B/row).
//   h (25.6MB) and out (25.6MB) both fit in 192MB L2 -> gather + f32 atomics
//   are L2-resident; HBM only streams the 32MB edge lists.
// ---------------------------------------------------------------------------
__global__ void __launch_bounds__(256)
gcn_spmm_scatter(const float* __restrict__ h, const float* __restrict__ vals,
                 const long long* __restrict__ rows,
                 const long long* __restrict__ cols,
                 float* __restrict__ out, int n_edges) {
  const int wid  = (blockIdx.x * 256 + (int)threadIdx.x) >> 5;
  const int lane = threadIdx.x & 31;
  int e = wid * EPW;
  if (e >= n_edges) return;
  const int e_end = (e + EPW < n_edges) ? (e + EPW) : n_edges;

  long long c = cols[e];
  for (; e < e_end; ++e) {
    const long long r = rows[e];
    const float     v = vals[e];
    const float4    m = *((const float4*)(h + (size_t)c * OUT_F) + lane);
    if (e + 1 < e_end) {
      const long long cn = cols[e + 1];
      __builtin_prefetch((const float4*)(h + (size_t)cn * OUT_F) + lane, 0, 1);
      c = cn;
    }
    float* op = out + (size_t)r * OUT_F + (lane << 2);
    atomicAdd(op + 0, v * m.x);
    atomicAdd(op + 1, v * m.y);
    atomicAdd(op + 2, v * m.z);
    atomicAdd(op + 3, v * m.w);
  }
}

// ---------------------------------------------------------------------------
// Host launcher
// ---------------------------------------------------------------------------
extern "C" void kernel_launch(void* const* d_in, const int* in_sizes, int n_in,
                              void* d_out, int out_size, void* d_ws, size_t ws_size,
                              hipStream_t stream) {
  const float*     x        = (const float*)d_in[0];
  const float*     W        = (const float*)d_in[1];
  const float*     b        = (const float*)d_in[2];
  const float*     adj_vals = (const float*)d_in[3];
  const long long* adj_row  = (const long long*)d_in[4];  // int64 in reference
  const long long* adj_col  = (const long long*)d_in[5];
  float* out = (float*)d_out;
  float* h   = (float*)d_ws;            // 50000*128*4 = 25.6 MB scratch
  const int n_edges = in_sizes[3];

  // 1) zero the output (segment_sum target)
  {
    int nthreads = (out_size + 3) / 4;
    gcn_zero<<<(nthreads + 255) / 256, 256, 0, stream>>>(out, out_size);
  }

  // 2) dense projection with WMMA (3125 blocks x 256 threads)
  gcn_gemm_wmma<<<N_NODES / TILE_M, 256, 0, stream>>>(x, W, b, h);

  // 3) edge scatter with L2-resident f32 atomics
  {
    int waves  = (n_edges + EPW - 1) / EPW;
    int blocks = (waves + 7) / 8;       // 8 waves per 256-thread block
    gcn_spmm_scatter<<<blocks, 256, 0, stream>>>(h, adj_vals, adj_row, adj_col,
                                                 out, n_edges);
  }
}